// Net_1640677507710
// MI455X (gfx1250) — compile-verified
//
#include <hip/hip_runtime.h>

// Problem dims (fixed by the reference)
#define STU_NUM  20000
#define ITEM_NUM 8000
#define CONC     128
#define EMB      32
#define BATCH    128

typedef float v2f __attribute__((ext_vector_type(2)));
typedef float v8f __attribute__((ext_vector_type(8)));
typedef int   v4i __attribute__((ext_vector_type(4)));

// Fast sigmoid: v_exp_f32 + v_rcp_f32 (avoids the IEEE-exact division
// sequence v_div_scale/v_div_fmas/v_div_fixup — ~12 VALU ops per call).
__device__ __forceinline__ float sigmoidf(float x) {
#if defined(__AMDGCN__)
    return __builtin_amdgcn_rcpf(1.0f + __expf(-x));
#else
    return 1.0f / (1.0f + __expf(-x));
#endif
}

// ---------------------------------------------------------------------------
// Kernel A: item side.
//   item_conc = sigmoid(item_emb @ w1^T + b1)         [I, C] via WMMA f32 16x16x4
//   Qw[i,c]   = Q[i,c] * wp[c]                        (B-matrix for kernel C)
//   t2[i]     = sum_c Qw[i,c] * item_conc[i,c]        (LDS ds_add_f32 reduce)
//   qadj[i]   = sum_c Q[i,c] * gw_Q[c]
// One block = 16 items; 8 waves = 8 concept tiles of 16.
// ---------------------------------------------------------------------------
__global__ __launch_bounds__(256)
void item_side_kernel(const float* __restrict__ item_emb,
                      const float* __restrict__ w1,
                      const float* __restrict__ b1,
                      const float* __restrict__ Q,
                      const float* __restrict__ wp,
                      const float* __restrict__ gwQ,
                      float* __restrict__ Qw,
                      float* __restrict__ t2,
                      float* __restrict__ qadj)
{
    __shared__ float t2_l[16];
    __shared__ float qa_l[16];

    const int tid  = threadIdx.x;
    const int wave = tid >> 5;
    const int lane = tid & 31;
    const int half = lane >> 4;   // 0: lanes 0-15, 1: lanes 16-31
    const int ln   = lane & 15;
    const int itemBase = blockIdx.x * 16;
    const int c0 = wave * 16;
    const int n  = c0 + ln;       // concept column owned by this lane

    if (tid < 16) { t2_l[tid] = 0.0f; qa_l[tid] = 0.0f; }
    __syncthreads();

    // D = item_emb[16x32] * w1^T[32x16], K split into 8 steps of 4
    v8f acc = {};
#pragma unroll
    for (int kk = 0; kk < 8; ++kk) {
        const int k = kk * 4 + half * 2;
        v2f a, b;
        a.x = item_emb[(itemBase + ln) * EMB + k];       // A[m][k],   m = ln
        a.y = item_emb[(itemBase + ln) * EMB + k + 1];   // A[m][k+1]
        b.x = w1[n * EMB + k];                           // B[k][n]   = w1[n][k]
        b.y = w1[n * EMB + k + 1];                       // B[k+1][n]
        acc = __builtin_amdgcn_wmma_f32_16x16x4_f32(
            /*neg_a=*/false, a, /*neg_b=*/false, b,
            /*c_mod=*/(short)0, acc, /*reuse_a=*/false, /*reuse_b=*/false);
    }

    const float wpc  = wp[n];
    const float gq   = gwQ[n];
    const float bias = b1[n];
#pragma unroll
    for (int v = 0; v < 8; ++v) {
        const int m = v + half * 8;          // C/D layout: M = v + 8*half
        const int i = itemBase + m;
        const float conc = sigmoidf(acc[v] + bias);
        const float q    = Q[i * CONC + n];
        const float qw   = q * wpc;
        Qw[i * CONC + n] = qw;
        atomicAdd(&t2_l[m], qw * conc);      // ds_add_f32
        atomicAdd(&qa_l[m], q * gq);
    }
    __syncthreads();
    if (tid < 16) {
        t2[itemBase + tid]   = t2_l[tid];
        qadj[itemBase + tid] = qa_l[tid];
    }
}

// ---------------------------------------------------------------------------
// Kernel B: student side (tiny).
//   stu_conc = sigmoid(stu_emb[stu_list-1] @ w1^T + b1)   [B, C]
//   gs[b]    = gw_stu[stu_list[b]-1]
// 8 blocks x 16 batch rows; 8 waves = 8 concept tiles.
// ---------------------------------------------------------------------------
__global__ __launch_bounds__(256)
void stu_side_kernel(const int*   __restrict__ stu_list,
                     const float* __restrict__ stu_emb,
                     const float* __restrict__ w1,
                     const float* __restrict__ b1,
                     const float* __restrict__ gw_stu,
                     float* __restrict__ sc,
                     float* __restrict__ gs)
{
    const int tid  = threadIdx.x;
    const int wave = tid >> 5;
    const int lane = tid & 31;
    const int half = lane >> 4;
    const int ln   = lane & 15;
    const int mBase = blockIdx.x * 16;   // batch rows for this block
    const int n     = wave * 16 + ln;    // concept column

    const int row = stu_list[mBase + ln] - 1;   // gathered A row (per-lane M)

    v8f acc = {};
#pragma unroll
    for (int kk = 0; kk < 8; ++kk) {
        const int k = kk * 4 + half * 2;
        v2f a, b;
        a.x = stu_emb[row * EMB + k];
        a.y = stu_emb[row * EMB + k + 1];
        b.x = w1[n * EMB + k];
        b.y = w1[n * EMB + k + 1];
        acc = __builtin_amdgcn_wmma_f32_16x16x4_f32(
            false, a, false, b, (short)0, acc, false, false);
    }

    const float bias = b1[n];
#pragma unroll
    for (int v = 0; v < 8; ++v) {
        const int b_row = mBase + v + half * 8;
        sc[b_row * CONC + n] = sigmoidf(acc[v] + bias);
    }
    if (wave == 0 && lane < 16) {
        const int b_row = mBase + lane;
        gs[b_row] = gw_stu[stu_list[b_row] - 1];
    }
}

// ---------------------------------------------------------------------------
// Kernel C: fused output GEMM + epilogue.
//   term1[b,i] = sum_c stu_conc[b,c] * Qw[i,c]      (M=128, N=8000, K=128)
//   P  = sigmoid(disc[i]*(term1 - t2[i]) + bp)
//   g1 = sigmoid(gs[b] + qadj[i]); g2 = sigmoid(g1*w2 + b2)
//   out[b,i] = g2 + (1-g2)*P
// One block = one 16-item N tile; the 8 KB contiguous Qw tile is staged into
// LDS via the CDNA5 async point-to-point path (GLOBAL_LOAD_ASYNC_TO_LDS_B128,
// tracked by ASYNCcnt) — no VGPR round-trip. 8 waves = 8 batch M tiles of 16;
// 32 WMMA steps over K=128.
// ---------------------------------------------------------------------------
__global__ __launch_bounds__(256)
void out_kernel(const float* __restrict__ sc,
                const float* __restrict__ Qw,
                const float* __restrict__ t2,
                const float* __restrict__ qadj,
                const float* __restrict__ gs,
                const float* __restrict__ disc,
                const float* __restrict__ bp,
                const float* __restrict__ w2,
                const float* __restrict__ b2,
                float* __restrict__ out)
{
    __shared__ float Bs[16 * CONC];   // Qw tile, [n][k], 8 KB

    const int tid  = threadIdx.x;
    const int wave = tid >> 5;
    const int lane = tid & 31;
    const int half = lane >> 4;
    const int ln   = lane & 15;
    const int itemBase = blockIdx.x * 16;

    // 16 consecutive item rows of Qw are one contiguous 8 KB span.
#if defined(__gfx1250__) && __has_builtin(__builtin_amdgcn_global_load_async_to_lds_b128)
    {
        // Builtin signature (from hipcc diagnostic): (v4i AS1*, v4i AS3*, Ii, Ii)
        typedef __attribute__((address_space(1))) v4i* gv4ip;
        typedef __attribute__((address_space(3))) v4i* lv4ip;
        const float* srcf = Qw + (size_t)itemBase * CONC;
#pragma unroll
        for (int r = 0; r < 2; ++r) {
            const int e = (tid + r * 256) * 4;   // 16-byte granules
            __builtin_amdgcn_global_load_async_to_lds_b128(
                (gv4ip)(srcf + e), (lv4ip)(Bs + e),
                /*offset=*/0, /*cpol=*/0);
        }
#if __has_builtin(__builtin_amdgcn_s_wait_asynccnt)
        __builtin_amdgcn_s_wait_asynccnt(0);
#else
        asm volatile("s_wait_asynccnt 0x0" ::: "memory");
#endif
    }
#else
    {
        const float4* src = (const float4*)(Qw + (size_t)itemBase * CONC);
        float4*       dst = (float4*)Bs;
#pragma unroll
        for (int r = 0; r < 2; ++r)
            dst[tid + r * 256] = src[tid + r * 256];
    }
#endif
    __syncthreads();

    const int mBase = wave * 16;     // this wave's batch rows
    v8f acc = {};
#pragma unroll
    for (int kk = 0; kk < 32; ++kk) {
        const int k = kk * 4 + half * 2;
        v2f a, b;
        a.x = sc[(mBase + ln) * CONC + k];      // A[m][k], m = ln
        a.y = sc[(mBase + ln) * CONC + k + 1];
        b.x = Bs[ln * CONC + k];                // B[k][n] = Qw[n][k], n = ln
        b.y = Bs[ln * CONC + k + 1];
        acc = __builtin_amdgcn_wmma_f32_16x16x4_f32(
            false, a, false, b, (short)0, acc, false, false);
    }

    const int   i   = itemBase + ln;
    const float di  = disc[i];
    const float t2i = t2[i];
    const float qi  = qadj[i];
    const float bp0 = bp[0];
    const float w20 = w2[0];
    const float b20 = b2[0];
#pragma unroll
    for (int v = 0; v < 8; ++v) {
        const int b_row = mBase + v + half * 8;
        const float P  = sigmoidf(di * (acc[v] - t2i) + bp0);
        const float g1 = sigmoidf(gs[b_row] + qi);
        const float g2 = sigmoidf(g1 * w20 + b20);
        out[(size_t)b_row * ITEM_NUM + i] = g2 + (1.0f - g2) * P;
    }
}

// ---------------------------------------------------------------------------
extern "C" void kernel_launch(void* const* d_in, const int* in_sizes, int n_in,
                              void* d_out, int out_size, void* d_ws, size_t ws_size,
                              hipStream_t stream) {
    (void)in_sizes; (void)n_in; (void)out_size; (void)ws_size;

    const int*   stu_list = (const int*)  d_in[0];
    const float* stu_emb  = (const float*)d_in[1];
    const float* item_emb = (const float*)d_in[2];
    const float* disc     = (const float*)d_in[3];
    const float* Q        = (const float*)d_in[4];
    const float* w1       = (const float*)d_in[5];
    const float* b1       = (const float*)d_in[6];
    const float* wp       = (const float*)d_in[7];
    const float* bp       = (const float*)d_in[8];
    const float* gw_stu   = (const float*)d_in[9];
    const float* gw_Q     = (const float*)d_in[10];
    const float* w2       = (const float*)d_in[11];
    const float* b2       = (const float*)d_in[12];
    float* out = (float*)d_out;

    // Workspace layout (floats): Qw | t2 | qadj | stu_conc | gs  (~4.03 MB)
    float* ws   = (float*)d_ws;
    float* Qw   = ws;
    float* t2   = Qw   + (size_t)ITEM_NUM * CONC;
    float* qadj = t2   + ITEM_NUM;
    float* sc   = qadj + ITEM_NUM;
    float* gs   = sc   + (size_t)BATCH * CONC;

    dim3 blk(256);
    item_side_kernel<<<ITEM_NUM / 16, blk, 0, stream>>>(
        item_emb, w1, b1, Q, wp, gw_Q, Qw, t2, qadj);
    stu_side_kernel<<<BATCH / 16, blk, 0, stream>>>(
        stu_list, stu_emb, w1, b1, gw_stu, sc, gs);
    out_kernel<<<ITEM_NUM / 16, blk, 0, stream>>>(
        sc, Qw, t2, qadj, gs, disc, bp, w2, b2, out);
}